// InstanceLoss_14310831030757
// MI455X (gfx1250) — compile-verified
//
#include <hip/hip_runtime.h>
#include <hip/hip_bf16.h>
#include <math.h>

typedef __attribute__((ext_vector_type(2))) float v2f;
typedef __attribute__((ext_vector_type(8))) float v8f;

#define PXCOUNT (512 * 512)         // P = 262144 pixels
#define NW 512                      // total waves per streaming pass
#define PIX_PER_WAVE (PXCOUNT / NW) // 512
#define NBLK (NW / 8)               // 256 threads/block = 8 waves

// workspace layout (float element offsets)
#define WS0_STRIDE 1056                 // per-wave pass-1 partial: pcs[16*64]+cnt[16]+ssq[16]
#define WS1_OFF   (NW * WS0_STRIDE)
#define WS1_PCS   (WS1_OFF + 0)         // 1024  summed per_chan_sum
#define WS1_CNT   (WS1_OFF + 1024)      // 16    summed counts
#define WS1_SSQ   (WS1_OFF + 1040)      // 16    summed sum-of-squares
#define WS1_MEANS (WS1_OFF + 1056)      // 1024
#define WS1_NORM  (WS1_OFF + 2080)      // 16
#define WS1_STD   (WS1_OFF + 2096)      // 16
#define WS2_OFF   (WS1_OFF + 2112)      // pass-3 partials: NW * 256

// ---- FP32 WMMA 16x16x4: D = A(16x4) * B(4x16) + C ----
__device__ __forceinline__ v8f wmma_f32_16x16x4(v2f a, v2f b, v8f c) {
#if __has_builtin(__builtin_amdgcn_wmma_f32_16x16x4_f32)
  return __builtin_amdgcn_wmma_f32_16x16x4_f32(false, a, false, b, (short)0, c,
                                               false, false);
#else
  // lane-exact emulation of the same register layout (compile-safe fallback)
  const int lane = threadIdx.x & 31;
  const int l15 = lane & 15, h = lane >> 4;
  float b0 = __shfl(b.x, l15, 32);
  float b1 = __shfl(b.y, l15, 32);
  float b2 = __shfl(b.x, l15 + 16, 32);
  float b3 = __shfl(b.y, l15 + 16, 32);
#pragma unroll
  for (int r = 0; r < 8; ++r) {
    int m = r + 8 * h;
    float a0 = __shfl(a.x, m, 32);
    float a1 = __shfl(a.y, m, 32);
    float a2 = __shfl(a.x, m + 16, 32);
    float a3 = __shfl(a.y, m + 16, 32);
    c[r] += a0 * b0 + a1 * b1 + a2 * b2 + a3 * b3;
  }
  return c;
#endif
}

// ------------------------------------------------------------------
// Pass 1: per_chan_sum = M @ X1^T, counts, sumsq  (stream views_1 once)
// 8 pixels/iter via B128 loads; K-slot mapping {p0,p1|p4,p5}, {p2,p3|p6,p7}.
// sumsq is computed as a 5th WMMA (B = per-lane channel-group squares),
// so the hot loop has ZERO cross-lane ops.
// ------------------------------------------------------------------
__global__ __launch_bounds__(256) void k1_stats(const float* __restrict__ X1,
                                                const int* __restrict__ Mi,
                                                float* __restrict__ ws) {
  const int lane = threadIdx.x & 31;
  const int wave = (blockIdx.x * 256 + threadIdx.x) >> 5;
  const int l15 = lane & 15;
  const int h = lane >> 4;
  const long pbase = (long)wave * PIX_PER_WAVE;

  v8f acc0 = {}, acc1 = {}, acc2 = {}, acc3 = {}, accS = {};
  float cnt_acc = 0.f;

  for (int it = 0; it < PIX_PER_WAVE; it += 8) {
    const long p0 = pbase + it;
    // masks: 4 pixels of this half in one B128
    const int4 m4 = *(const int4*)(Mi + (long)l15 * PXCOUNT + p0 + 4 * h);
    const v2f Aa = {(float)m4.x, (float)m4.y};  // K = {p0,p0+1 | p0+4,p0+5}
    const v2f Ab = {(float)m4.z, (float)m4.w};  // K = {p0+2,p0+3 | p0+6,p0+7}
    // X1: 4 channel tiles x 4 pixels of this half, B128 each
    const float4 b0 = *(const float4*)(X1 + (long)(l15 + 0)  * PXCOUNT + p0 + 4 * h);
    const float4 b1 = *(const float4*)(X1 + (long)(l15 + 16) * PXCOUNT + p0 + 4 * h);
    const float4 b2 = *(const float4*)(X1 + (long)(l15 + 32) * PXCOUNT + p0 + 4 * h);
    const float4 b3 = *(const float4*)(X1 + (long)(l15 + 48) * PXCOUNT + p0 + 4 * h);
    // pixel group a
    acc0 = wmma_f32_16x16x4(Aa, (v2f){b0.x, b0.y}, acc0);
    acc1 = wmma_f32_16x16x4(Aa, (v2f){b1.x, b1.y}, acc1);
    acc2 = wmma_f32_16x16x4(Aa, (v2f){b2.x, b2.y}, acc2);
    acc3 = wmma_f32_16x16x4(Aa, (v2f){b3.x, b3.y}, acc3);
    // pixel group b
    acc0 = wmma_f32_16x16x4(Ab, (v2f){b0.z, b0.w}, acc0);
    acc1 = wmma_f32_16x16x4(Ab, (v2f){b1.z, b1.w}, acc1);
    acc2 = wmma_f32_16x16x4(Ab, (v2f){b2.z, b2.w}, acc2);
    acc3 = wmma_f32_16x16x4(Ab, (v2f){b3.z, b3.w}, acc3);
    // per-(pixel, channel-group) partial squares -> sumsq WMMA
    const float px = b0.x * b0.x + b1.x * b1.x + b2.x * b2.x + b3.x * b3.x;
    const float py = b0.y * b0.y + b1.y * b1.y + b2.y * b2.y + b3.y * b3.y;
    const float pz = b0.z * b0.z + b1.z * b1.z + b2.z * b2.z + b3.z * b3.z;
    const float pw = b0.w * b0.w + b1.w * b1.w + b2.w * b2.w + b3.w * b3.w;
    accS = wmma_f32_16x16x4(Aa, (v2f){px, py}, accS);
    accS = wmma_f32_16x16x4(Ab, (v2f){pz, pw}, accS);
    cnt_acc += Aa.x + Aa.y + Ab.x + Ab.y;
  }

  float* o = ws + (long)wave * WS0_STRIDE;
#pragma unroll
  for (int r = 0; r < 8; ++r) {  // D layout: row m = r+8h, col = channel-in-tile
    const int m = r + 8 * h;
    o[m * 64 + 0  + l15] = acc0[r];
    o[m * 64 + 16 + l15] = acc1[r];
    o[m * 64 + 32 + l15] = acc2[r];
    o[m * 64 + 48 + l15] = acc3[r];
  }
  // accS columns (channel-groups) -> sum within each 16-lane half, once per wave
#pragma unroll
  for (int r = 0; r < 8; ++r) {
    float v = accS[r];
#pragma unroll
    for (int d = 1; d < 16; d <<= 1) v += __shfl_xor(v, d, 32);
    if (l15 == 0) o[1040 + r + 8 * h] = v;  // row m = r+8h
  }
  cnt_acc += __shfl_xor(cnt_acc, 16, 32);
  if (lane < 16) o[1024 + lane] = cnt_acc;
}

// ------------------------------------------------------------------
// Pass 2: reduce partials; means, norm_m, std
// ------------------------------------------------------------------
__global__ __launch_bounds__(256) void k2_finalize(float* __restrict__ ws) {
  const int tid = threadIdx.x;
  for (int idx = tid; idx < WS0_STRIDE; idx += 256) {
    float s = 0.f;
    for (int w = 0; w < NW; ++w) s += ws[(long)w * WS0_STRIDE + idx];
    ws[WS1_OFF + idx] = s;
  }
  __syncthreads();
  for (int idx = tid; idx < 1024; idx += 256) {
    const int k = idx >> 6;
    const float c1 = fmaxf(ws[WS1_CNT + k], 1.f);
    ws[WS1_MEANS + idx] = ws[WS1_PCS + idx] / c1;
  }
  __syncthreads();
  if (tid < 16) {
    const int k = tid;
    const float cnt = ws[WS1_CNT + k];
    const float sumsq = ws[WS1_SSQ + k];
    float nsq = 0.f, sum_all = 0.f;
    for (int c = 0; c < 64; ++c) {
      const float m = ws[WS1_MEANS + k * 64 + c];
      nsq += m * m;
      sum_all += ws[WS1_PCS + k * 64 + c];
    }
    ws[WS1_NORM + k] = sqrtf(nsq);
    const float n_elem = cnt * 64.f;
    const float mean_all = sum_all / fmaxf(n_elem, 1.f);
    const float var =
        (sumsq - n_elem * mean_all * mean_all) / fmaxf(n_elem - 1.f, 1.f);
    ws[WS1_STD + k] = sqrtf(fmaxf(var, 0.f));
  }
}

// ------------------------------------------------------------------
// Pass 3: D^T = X2^T @ means^T; ratio; simT += M @ ratio^T (stream views_2)
// Second-WMMA chunks use K = {2c,2c+1 | 2c+8,2c+9}: the ratio registers in
// first-WMMA D layout ARE the B operands (zero shuffles); masks come from
// two B128 loads per 16-pixel tile.
// ------------------------------------------------------------------
__global__ __launch_bounds__(256) void k3_sim(const float* __restrict__ X2,
                                              const int* __restrict__ Mi,
                                              float* __restrict__ ws) {
  const int lane = threadIdx.x & 31;
  const int wave = (blockIdx.x * 256 + threadIdx.x) >> 5;
  const int l15 = lane & 15;
  const int h = lane >> 4;
  const long pbase = (long)wave * PIX_PER_WAVE;
  const float* means = ws + WS1_MEANS;

  // loop-invariant B operand: means^T chunks, K = 4 channels each
  v2f Bm[16];
#pragma unroll
  for (int kk = 0; kk < 16; ++kk) {
    const float2 mv = *(const float2*)(means + l15 * 64 + kk * 4 + 2 * h);
    Bm[kk] = (v2f){mv.x, mv.y};
  }
  const float nm = ws[WS1_NORM + l15];  // norm_m for column (mean index) = l15

  v8f sim = {};

  for (int t = 0; t < PIX_PER_WAVE; t += 16) {
    const long p = pbase + t + l15;  // A row = pixel-in-tile = l15
    v8f D = {};
    float nsq = 0.f;
#pragma unroll
    for (int kk = 0; kk < 16; ++kk) {
      const int c0 = kk * 4 + 2 * h;
      const float a0 = X2[(long)c0 * PXCOUNT + p];
      const float a1 = X2[(long)(c0 + 1) * PXCOUNT + p];
      D = wmma_f32_16x16x4((v2f){a0, a1}, Bm[kk], D);
      nsq += a0 * a0 + a1 * a1;
    }
    nsq += __shfl_xor(nsq, 16, 32);  // full ||X2[:,p]||^2 at lane = pixel

    // ratio^T[p][i] = D^T / max(norm_m[i] * norm2[p], eps); row pixel = r+8h
    v8f R;
#pragma unroll
    for (int r = 0; r < 8; ++r) {
      const float n2 = __shfl(nsq, r + 8 * h, 32);
      const float denom = fmaxf(nm * sqrtf(n2), 1e-8f);
      R[r] = D[r] / denom;
    }

    // masks for all 4 chunks: two B128 loads
    const int4 ma = *(const int4*)(Mi + (long)l15 * PXCOUNT + pbase + t + 8 * h);
    const int4 mb = *(const int4*)(Mi + (long)l15 * PXCOUNT + pbase + t + 4 + 8 * h);
    // chunk c: K = {2c, 2c+1 | 2c+8, 2c+9}; B = (R[2c], R[2c+1]) in place
    sim = wmma_f32_16x16x4((v2f){(float)ma.x, (float)ma.y}, (v2f){R[0], R[1]}, sim);
    sim = wmma_f32_16x16x4((v2f){(float)ma.z, (float)ma.w}, (v2f){R[2], R[3]}, sim);
    sim = wmma_f32_16x16x4((v2f){(float)mb.x, (float)mb.y}, (v2f){R[4], R[5]}, sim);
    sim = wmma_f32_16x16x4((v2f){(float)mb.z, (float)mb.w}, (v2f){R[6], R[7]}, sim);
  }

  float* o = ws + WS2_OFF + (long)wave * 256;
#pragma unroll
  for (int r = 0; r < 8; ++r)  // store simT[j = r+8h][i = l15]
    o[(r + 8 * h) * 16 + l15] = sim[r];
}

// ------------------------------------------------------------------
// Pass 4: reduce sim, label scatter, buckets, final 36 outputs
// ------------------------------------------------------------------
__global__ __launch_bounds__(256) void k4_final(const int* __restrict__ labels,
                                                const float* __restrict__ ws,
                                                float* __restrict__ out) {
  __shared__ float simS[16][16];  // simS[i][j], i = mean row, j = mask col
  const int tid = threadIdx.x;
  {
    const int j = tid >> 4, i = tid & 15;
    float s = 0.f;
    for (int w = 0; w < NW; ++w) s += ws[WS2_OFF + (long)w * 256 + tid];
    const float cnt = fmaxf(ws[WS1_CNT + j], 1.f);
    simS[i][j] = s / cnt;
  }
  __syncthreads();
  if (tid == 0) {
    int lab[16];
    for (int k = 0; k < 16; ++k) lab[k] = labels[k];
    float inst[11] = {}, cls[11] = {}, cstd[11] = {}, ccnt[11] = {};
    for (int k = 0; k < 16; ++k) {
      inst[lab[k]] += simS[k][k];
      cstd[lab[k]] += ws[WS1_STD + k];
      ccnt[lab[k]] += 1.f;
      float s = 0.f;
      for (int j = 0; j < 16; ++j)
        if (j != k && lab[j] == lab[k]) s += simS[k][j];
      cls[lab[k]] += s;
    }
    float s1 = 0, c1 = 0, s2 = 0, c2 = 0, s3 = 0, c3 = 0;
    for (int a = 0; a < 16; ++a)
      for (int b = 0; b < 16; ++b) {
        if (lab[a] == lab[b]) continue;
        const int la = lab[a], lb = lab[b];
        const bool ga1 = la >= 1 && la <= 8, gb1 = lb >= 1 && lb <= 8;
        const bool ga2 = la >= 3 && la <= 10, gb2 = lb >= 3 && lb <= 10;
        const bool ga3 = (la >= 1 && la <= 2) || (la >= 9 && la <= 10);
        const bool gb3 = (lb >= 1 && lb <= 2) || (lb >= 9 && lb <= 10);
        const float v = simS[a][b];
        if (ga1 && gb1) { s1 += v; c1 += 1.f; }
        if (ga2 && gb2) { s2 += v; c2 += 1.f; }
        if (ga3 && gb3) { s3 += v; c3 += 1.f; }
      }
    for (int c = 0; c < 11; ++c) {
      const float cc = ccnt[c];
      const bool many = cc > 1.f;
      out[c]      = many ? inst[c] / fmaxf(cc, 1.f) : inst[c];
      out[11 + c] = many ? cls[c] / fmaxf(cc * (cc - 1.f), 1.f) : cls[c];
      out[25 + c] = many ? cstd[c] / fmaxf(cc, 1.f) : cstd[c];
    }
    out[22] = (c1 > 1.f) ? s1 / c1 : s1;
    out[23] = (c2 > 1.f) ? s2 / c2 : s2;
    out[24] = (c3 > 1.f) ? s3 / c3 : s3;
  }
}

extern "C" void kernel_launch(void* const* d_in, const int* in_sizes, int n_in,
                              void* d_out, int out_size, void* d_ws,
                              size_t ws_size, hipStream_t stream) {
  const float* X1 = (const float*)d_in[0];   // views_1 [64, 512*512]
  const float* X2 = (const float*)d_in[1];   // views_2 [64, 512*512]
  const int* Mi = (const int*)d_in[2];       // masks   [16, 512*512]
  const int* lab = (const int*)d_in[3];      // labels  [16]
  float* out = (float*)d_out;                // 36 floats
  float* ws = (float*)d_ws;

  k1_stats<<<NBLK, 256, 0, stream>>>(X1, Mi, ws);
  k2_finalize<<<1, 256, 0, stream>>>(ws);
  k3_sim<<<NBLK, 256, 0, stream>>>(X2, Mi, ws);
  k4_final<<<1, 256, 0, stream>>>(lab, ws, out);
}